// CLUB_NCE_1434519077323
// MI455X (gfx1250) — compile-verified
//
#include <hip/hip_runtime.h>
#include <math.h>

typedef __attribute__((ext_vector_type(2))) float v2f;
typedef __attribute__((ext_vector_type(8))) float v8f;

#define D_DIM 512
#define N_SAMP 1024

// out[M x 512] = In[M x 512] @ W[512 x 512] (+ optional bias per column)
// One wave computes a 16x32 C tile (two 16x16 WMMA accumulators, shared A frag).
// A-frag (16x4 f32): lane l -> row M = l&15, VGPR v -> K = 2*(l>>4)+v  (contiguous pair -> b64 load)
// B-frag (4x16 f32): lane l -> col N = l&15, VGPR v -> K = 2*(l>>4)+v
// C/D (16x16 f32):  lane l -> N = l&15, VGPR r -> M = r + 8*(l>>4)
__global__ __launch_bounds__(256) void club_gemm_wmma(
    const float* __restrict__ In, const float* __restrict__ W,
    const float* __restrict__ bias, int has_bias, float* __restrict__ out)
{
    const int K = D_DIM, Nn = D_DIM;
    const int tid   = threadIdx.x;
    const int lane  = tid & 31;
    const int wave  = tid >> 5;
    const int w     = blockIdx.x * 8 + wave;      // global wave id
    const int tiles_n = Nn / 32;                  // 16 tile-columns of width 32
    const int tn  = (w % tiles_n) * 32;
    const int tm  = (w / tiles_n) * 16;
    const int half  = lane >> 4;
    const int idx16 = lane & 15;

    const float* arow  = In + (size_t)(tm + idx16) * K + 2 * half;
    const float* bcol0 = W + (size_t)(2 * half) * Nn + tn + idx16;
    const float* bcol1 = bcol0 + 16;

    v8f acc0 = {};
    v8f acc1 = {};
    for (int k0 = 0; k0 < K; k0 += 4) {
        v2f a = *(const v2f*)(arow + k0);              // K = k0+2h, k0+2h+1
        v2f b0, b1v;
        b0.x  = bcol0[(size_t)(k0    ) * Nn];
        b0.y  = bcol0[(size_t)(k0 + 1) * Nn];
        b1v.x = bcol1[(size_t)(k0    ) * Nn];
        b1v.y = bcol1[(size_t)(k0 + 1) * Nn];
        acc0 = __builtin_amdgcn_wmma_f32_16x16x4_f32(false, a, false, b0,
                                                     (short)0, acc0, false, false);
        acc1 = __builtin_amdgcn_wmma_f32_16x16x4_f32(false, a, false, b1v,
                                                     (short)0, acc1, false, false);
    }

    const float bv0 = has_bias ? bias[tn + idx16]      : 0.0f;
    const float bv1 = has_bias ? bias[tn + 16 + idx16] : 0.0f;
#pragma unroll
    for (int r = 0; r < 8; ++r) {
        const int row = tm + half * 8 + r;
        out[(size_t)row * Nn + tn + idx16]      = acc0[r] + bv0;
        out[(size_t)row * Nn + tn + 16 + idx16] = acc1[r] + bv1;
    }
}

// One block per row a: T1[a,b] = softplus(dot(relu(xpb[b]+yp[a]), W2) + b2)
// Online logsumexp over b, row-sum, and diagonal (T0) capture.
__global__ __launch_bounds__(256) void club_pairwise(
    const float* __restrict__ xpb, const float* __restrict__ yp,
    const float* __restrict__ W2,  const float* __restrict__ b2,
    float* __restrict__ lse, float* __restrict__ rowsum, float* __restrict__ diag)
{
    __shared__ float s_y[D_DIM];
    __shared__ float s_w[D_DIM];
    __shared__ float s_m[8], s_s[8], s_sum[8], s_d[8];

    const int i   = blockIdx.x;
    const int tid = threadIdx.x;

    s_y[tid]       = yp[(size_t)i * D_DIM + tid];
    s_y[tid + 256] = yp[(size_t)i * D_DIM + tid + 256];
    s_w[tid]       = W2[tid];
    s_w[tid + 256] = W2[tid + 256];
    __syncthreads();

    const int lane = tid & 31;
    const int wave = tid >> 5;
    const float b2v = b2[0];

    float m = -3.0e38f, s = 0.0f, asum = 0.0f, dval = 0.0f;

    for (int j = wave; j < N_SAMP; j += 8) {
        const float* xr = xpb + (size_t)j * D_DIM;
        if (j + 8 < N_SAMP)  // stream-ahead hint -> global_prefetch_b8
            __builtin_prefetch(xpb + (size_t)(j + 8) * D_DIM + lane * 16, 0, 1);

        float dot = 0.0f;
#pragma unroll
        for (int t = 0; t < D_DIM / 32; ++t) {
            const int k = lane + t * 32;
            float h = xr[k] + s_y[k];
            h = fmaxf(h, 0.0f);
            dot = fmaf(h, s_w[k], dot);
        }
        // wave32 tree reduction
        for (int off = 16; off > 0; off >>= 1)
            dot += __shfl_down(dot, off, 32);

        if (lane == 0) {
            const float z  = dot + b2v;
            const float t1 = fmaxf(z, 0.0f) + log1pf(expf(-fabsf(z)));  // softplus
            const float mn = fmaxf(m, t1);
            s = s * expf(m - mn) + expf(t1 - mn);
            m = mn;
            asum += t1;
            if (j == i) dval = t1;   // T0[i] == T1[i,i]
        }
    }

    if (lane == 0) { s_m[wave] = m; s_s[wave] = s; s_sum[wave] = asum; s_d[wave] = dval; }
    __syncthreads();

    if (tid == 0) {
        float mm = s_m[0];
        for (int wv = 1; wv < 8; ++wv) mm = fmaxf(mm, s_m[wv]);
        float S = 0.0f, rs = 0.0f, dd = 0.0f;
        for (int wv = 0; wv < 8; ++wv) {
            S  += s_s[wv] * expf(s_m[wv] - mm);
            rs += s_sum[wv];
            dd += s_d[wv];
        }
        lse[i]    = mm + logf(S);
        rowsum[i] = rs;
        diag[i]   = dd;
    }
}

__global__ __launch_bounds__(256) void club_finalize(
    const float* __restrict__ lse, const float* __restrict__ rowsum,
    const float* __restrict__ diag, float* __restrict__ out)
{
    __shared__ float r0[256], r1[256], r2[256];
    const int tid = threadIdx.x;
    float a = 0.0f, b = 0.0f, c = 0.0f;
    for (int k = tid; k < N_SAMP; k += 256) {
        a += lse[k]; b += rowsum[k]; c += diag[k];
    }
    r0[tid] = a; r1[tid] = b; r2[tid] = c;
    __syncthreads();
    for (int off = 128; off > 0; off >>= 1) {
        if (tid < off) {
            r0[tid] += r0[tid + off];
            r1[tid] += r1[tid + off];
            r2[tid] += r2[tid + off];
        }
        __syncthreads();
    }
    if (tid == 0) {
        const float Nf  = (float)N_SAMP;
        const float t0m = r2[0] / Nf;                         // mean(diag) == T0.mean()
        out[0] = t0m - (r0[0] / Nf - logf(Nf));               // lower bound
        out[1] = t0m - r1[0] / (Nf * Nf);                     // upper bound
    }
}

extern "C" void kernel_launch(void* const* d_in, const int* in_sizes, int n_in,
                              void* d_out, int out_size, void* d_ws, size_t ws_size,
                              hipStream_t stream) {
    const float* x  = (const float*)d_in[0];   // (1024, 512)
    const float* y  = (const float*)d_in[1];   // (1024, 512)
    const float* W1 = (const float*)d_in[2];   // (1024, 512) row-major
    const float* b1 = (const float*)d_in[3];   // (512,)
    const float* W2 = (const float*)d_in[4];   // (512, 1) -> flat 512
    const float* b2 = (const float*)d_in[5];   // (1,)
    float* out = (float*)d_out;

    char*  ws     = (char*)d_ws;
    float* xpb    = (float*)(ws);                                 // 1024*512 f32 (= xp + b1)
    float* ypb    = (float*)(ws + (size_t)N_SAMP * D_DIM * 4);    // 1024*512 f32
    float* lse    = (float*)(ws + 2 * (size_t)N_SAMP * D_DIM * 4);
    float* rowsum = lse + N_SAMP;
    float* diag   = rowsum + N_SAMP;

    // 1024 waves per GEMM: (1024/16) x (512/32) tiles, 8 waves/block -> 128 blocks
    club_gemm_wmma<<<128, 256, 0, stream>>>(x, W1,                 b1, 1, xpb);
    club_gemm_wmma<<<128, 256, 0, stream>>>(y, W1 + (size_t)D_DIM * D_DIM, b1, 0, ypb);

    club_pairwise<<<N_SAMP, 256, 0, stream>>>(xpb, ypb, W2, b2, lse, rowsum, diag);
    club_finalize<<<1, 256, 0, stream>>>(lse, rowsum, diag, out);
}